// new_loss_13915694039429
// MI455X (gfx1250) — compile-verified
//
#include <hip/hip_runtime.h>
#include <math.h>

typedef float v2f __attribute__((ext_vector_type(2)));
typedef float v8f __attribute__((ext_vector_type(8)));

#define P 4096
#define NPART 4096   // partial-sum slots for both reductions

__device__ __forceinline__ v8f wmma_f32(v2f a, v2f b, v8f c) {
  return __builtin_amdgcn_wmma_f32_16x16x4_f32(false, a, false, b, (short)0, c,
                                               false, false);
}

// ---------------------------------------------------------------------------
// GEMM core: one wave computes a 64x64 fp32 tile of C = A * B,
// K swept 4 at a time with V_WMMA_F32_16X16X4_F32, explicit ping-pong double
// buffering driven by running pointers (advanced by a constant per trip) so
// the loop-top loads have no same-trip address dependency and can be clause-
// issued before the first WMMA group.
//  A fragment (16x4, f32): lane L holds M = L%16, K = vgpr + 2*(L/16)
//  B fragment (4x16, f32): lane L holds N = L%16, K = vgpr + 2*(L/16)
//  C/D (16x16, f32, 8 VGPRs): vgpr r holds M = r + 8*(L/16), N = L%16
// TRANSB=true means B[k][n] = Bmem[n*P + k] (i.e. B = X^T for G = Y*X^T).
// Fragment-row deltas (t*16*P elements, +16B for the odd k-step) become
// immediate instruction offsets.
// ---------------------------------------------------------------------------
template <bool TRANSB>
__device__ __forceinline__ void load_frag(const float* __restrict__ pa,
                                          const float* __restrict__ pb,
                                          int k, v2f a[4], v2f b[4]) {
#pragma unroll
  for (int t = 0; t < 4; ++t) {
    a[t] = *(const v2f*)(pa + t * 16 * P + k);
    if (TRANSB) {
      b[t] = *(const v2f*)(pb + t * 16 * P + k);
    } else {
      b[t].x = pb[t * 16 + k * P];
      b[t].y = pb[t * 16 + k * P + P];
    }
  }
}

__device__ __forceinline__ void do_wmma(const v2f a[4], const v2f b[4],
                                        v8f acc[4][4]) {
#pragma unroll
  for (int mt = 0; mt < 4; ++mt)
#pragma unroll
    for (int nt = 0; nt < 4; ++nt)
      acc[mt][nt] = wmma_f32(a[mt], b[nt], acc[mt][nt]);
}

template <bool TRANSB>
__device__ __forceinline__ void gemm_core(const float* __restrict__ A,
                                          const float* __restrict__ B,
                                          int m0, int n0, int lm, int kh,
                                          v8f acc[4][4]) {
  const float* pa = A + (m0 + lm) * P + 2 * kh;
  const float* pb = TRANSB ? (B + (n0 + lm) * P + 2 * kh)
                           : (B + 2 * kh * P + n0 + lm);

  v2f a0[4], b0[4], a1[4], b1[4];
  // prologue: k-step 0 into buffer 0
  load_frag<TRANSB>(pa, pb, 0, a0, b0);

  // Trip it: computes steps 2it and 2it+1, leaves step 2it+2 in flight.
#pragma unroll 1
  for (int it = 0; it < P / 8 - 1; ++it) {
    load_frag<TRANSB>(pa, pb, 4, a1, b1);   // step 2it+1 (no same-trip deps)
    do_wmma(a0, b0, acc);                   // step 2it
    pa += 8;
    pb += TRANSB ? 8 : 8 * P;
    load_frag<TRANSB>(pa, pb, 0, a0, b0);   // step 2it+2
    do_wmma(a1, b1, acc);                   // step 2it+1
  }
  // tail: buffer 0 holds step P/4-2; load final step P/4-1 into buffer 1
  load_frag<TRANSB>(pa, pb, 4, a1, b1);
  do_wmma(a0, b0, acc);
  do_wmma(a1, b1, acc);
}

// Block = 256 threads = 8 waves; block tile M=128 (2 waves) x N=256 (4 waves).
// grid = dim3(P/256, P/128).
__device__ __forceinline__ void tile_coords(int& m0, int& n0, int& lm, int& kh,
                                            int& wave) {
  int tid = threadIdx.x;
  wave = tid >> 5;
  int lane = tid & 31;
  lm = lane & 15;
  kh = lane >> 4;
  int wm = wave & 1;   // 0..1 along M
  int wn = wave >> 1;  // 0..3 along N
  m0 = blockIdx.y * 128 + wm * 64;
  n0 = blockIdx.x * 256 + wn * 64;
}

__device__ __forceinline__ void zero_acc(v8f acc[4][4]) {
#pragma unroll
  for (int mt = 0; mt < 4; ++mt)
#pragma unroll
    for (int nt = 0; nt < 4; ++nt)
#pragma unroll
      for (int r = 0; r < 8; ++r) acc[mt][nt][r] = 0.0f;
}

// GEMM1: W_matrix = scale[i] * (Y @ X^T)   -> out (d_out+1)
__global__ __launch_bounds__(256) void k_gemm1(const float* __restrict__ Y,
                                               const float* __restrict__ X,
                                               const float* __restrict__ scale,
                                               float* __restrict__ Wm) {
  int m0, n0, lm, kh, wave;
  tile_coords(m0, n0, lm, kh, wave);
  v8f acc[4][4];
  zero_acc(acc);

  gemm_core<true>(Y, X, m0, n0, lm, kh, acc);

#pragma unroll
  for (int mt = 0; mt < 4; ++mt)
#pragma unroll
    for (int r = 0; r < 8; ++r) {
      int row = m0 + mt * 16 + r + 8 * kh;
      float sc = scale[row];
#pragma unroll
      for (int nt = 0; nt < 4; ++nt) {
        int col = n0 + nt * 16 + lm;
        Wm[(size_t)row * P + col] = sc * acc[mt][nt][r];
      }
    }
}

// GEMM2: primary = W_matrix @ X - Y   -> ws.primary
__global__ __launch_bounds__(256) void k_gemm2(const float* __restrict__ Wm,
                                               const float* __restrict__ X,
                                               const float* __restrict__ Y,
                                               float* __restrict__ prim) {
  int m0, n0, lm, kh, wave;
  tile_coords(m0, n0, lm, kh, wave);
  v8f acc[4][4];
  zero_acc(acc);

  gemm_core<false>(Wm, X, m0, n0, lm, kh, acc);

#pragma unroll
  for (int mt = 0; mt < 4; ++mt)
#pragma unroll
    for (int r = 0; r < 8; ++r) {
      int row = m0 + mt * 16 + r + 8 * kh;
#pragma unroll
      for (int nt = 0; nt < 4; ++nt) {
        int col = n0 + nt * 16 + lm;
        size_t idx = (size_t)row * P + col;
        prim[idx] = acc[mt][nt][r] - Y[idx];
      }
    }
}

// GEMM3: Z = W_beta @ primary; per-wave partial of sum(Z^2) -> zpart (no atomics)
__global__ __launch_bounds__(256) void k_gemm3(const float* __restrict__ Wb,
                                               const float* __restrict__ prim,
                                               float* __restrict__ zpart) {
  int m0, n0, lm, kh, wave;
  tile_coords(m0, n0, lm, kh, wave);
  v8f acc[4][4];
  zero_acc(acc);

  gemm_core<false>(Wb, prim, m0, n0, lm, kh, acc);

  float local = 0.0f;
#pragma unroll
  for (int mt = 0; mt < 4; ++mt)
#pragma unroll
    for (int nt = 0; nt < 4; ++nt)
#pragma unroll
      for (int r = 0; r < 8; ++r) {
        float z = acc[mt][nt][r];
        local += z * z;
      }
  // wave32 reduce (fixed shuffle tree -> deterministic)
#pragma unroll
  for (int off = 16; off >= 1; off >>= 1) local += __shfl_xor(local, off, 32);
  if ((threadIdx.x & 31) == 0) {
    int bid = blockIdx.y * gridDim.x + blockIdx.x;
    zpart[bid * 8 + wave] = local;
  }
}

// ---------------------------------------------------------------------------
// Reductions (deterministic two-pass, fixed tree order)
// ---------------------------------------------------------------------------
__global__ __launch_bounds__(256) void k_sumsq(const float* __restrict__ X,
                                               float* __restrict__ spart) {
  __shared__ float sm[256];
  size_t base = (size_t)blockIdx.x * 4096 + threadIdx.x;
  float s = 0.0f;
#pragma unroll
  for (int j = 0; j < 16; ++j) {
    float v = X[base + (size_t)j * 256];
    s += v * v;
  }
  sm[threadIdx.x] = s;
  __syncthreads();
  for (int off = 128; off >= 1; off >>= 1) {
    if (threadIdx.x < off) sm[threadIdx.x] += sm[threadIdx.x + off];
    __syncthreads();
  }
  if (threadIdx.x == 0) spart[blockIdx.x] = sm[0];
}

__global__ __launch_bounds__(256) void k_reduce_s(const float* __restrict__ spart,
                                                  float* __restrict__ sval) {
  __shared__ float sm[256];
  float s = 0.0f;
#pragma unroll
  for (int j = 0; j < NPART / 256; ++j) s += spart[threadIdx.x + j * 256];
  sm[threadIdx.x] = s;
  __syncthreads();
  for (int off = 128; off >= 1; off >>= 1) {
    if (threadIdx.x < off) sm[threadIdx.x] += sm[threadIdx.x + off];
    __syncthreads();
  }
  if (threadIdx.x == 0) *sval = sm[0];
}

// One block per row of W_beta: W = rowsum; scale = W^2 / (s*W^2 + 1)
__global__ __launch_bounds__(256) void k_rowscale(const float* __restrict__ Wb,
                                                  const float* __restrict__ sval,
                                                  float* __restrict__ scale) {
  __shared__ float sm[256];
  size_t base = (size_t)blockIdx.x * P + threadIdx.x;
  float s = 0.0f;
#pragma unroll
  for (int j = 0; j < P / 256; ++j) s += Wb[base + (size_t)j * 256];
  sm[threadIdx.x] = s;
  __syncthreads();
  for (int off = 128; off >= 1; off >>= 1) {
    if (threadIdx.x < off) sm[threadIdx.x] += sm[threadIdx.x + off];
    __syncthreads();
  }
  if (threadIdx.x == 0) {
    float W = sm[0];
    float W2 = W * W;
    scale[blockIdx.x] = W2 / ((*sval) * W2 + 1.0f);
  }
}

__global__ __launch_bounds__(256) void k_final_loss(const float* __restrict__ zpart,
                                                    float* __restrict__ out0) {
  __shared__ float sm[256];
  float s = 0.0f;
#pragma unroll
  for (int j = 0; j < NPART / 256; ++j) s += zpart[threadIdx.x + j * 256];
  sm[threadIdx.x] = s;
  __syncthreads();
  for (int off = 128; off >= 1; off >>= 1) {
    if (threadIdx.x < off) sm[threadIdx.x] += sm[threadIdx.x + off];
    __syncthreads();
  }
  if (threadIdx.x == 0) *out0 = sqrtf(sm[0]);
}

// ---------------------------------------------------------------------------
// Launcher. d_in: [target(X), source(Y), W_beta]; d_out: [loss, W_matrix flat].
// ws layout (floats): primary[P*P] | spart[4096] | zpart[4096] | scale[P] | s[1]
// ---------------------------------------------------------------------------
extern "C" void kernel_launch(void* const* d_in, const int* in_sizes, int n_in,
                              void* d_out, int out_size, void* d_ws, size_t ws_size,
                              hipStream_t stream) {
  const float* X  = (const float*)d_in[0];  // target
  const float* Y  = (const float*)d_in[1];  // source
  const float* Wb = (const float*)d_in[2];  // W_beta

  float* out  = (float*)d_out;
  float* loss = out;      // out[0]
  float* Wm   = out + 1;  // out[1 .. 1+P*P)

  float* ws    = (float*)d_ws;
  float* prim  = ws;                         // P*P
  float* spart = ws + (size_t)P * P;         // NPART
  float* zpart = spart + NPART;              // NPART
  float* scale = zpart + NPART;              // P
  float* sval  = scale + P;                  // 1

  dim3 gemm_grid(P / 256, P / 128);
  dim3 gemm_block(256);

  // 1) s = ||X||_F^2  (two-pass deterministic)
  k_sumsq<<<dim3(NPART), dim3(256), 0, stream>>>(X, spart);
  k_reduce_s<<<dim3(1), dim3(256), 0, stream>>>(spart, sval);
  // 2) scale[i] = W_i^2 / (s*W_i^2 + 1)
  k_rowscale<<<dim3(P), dim3(256), 0, stream>>>(Wb, sval, scale);
  // 3) W_matrix = scale[:,None] * (Y @ X^T)
  k_gemm1<<<gemm_grid, gemm_block, 0, stream>>>(Y, X, scale, Wm);
  // 4) primary = W_matrix @ X - Y
  k_gemm2<<<gemm_grid, gemm_block, 0, stream>>>(Wm, X, Y, prim);
  // 5) Z = W_beta @ primary; per-wave sum(Z^2) partials
  k_gemm3<<<gemm_grid, gemm_block, 0, stream>>>(Wb, prim, zpart);
  // 6) loss = sqrt(sum partials)
  k_final_loss<<<dim3(1), dim3(256), 0, stream>>>(zpart, loss);
}